// lossFunc_11321533792493
// MI455X (gfx1250) — compile-verified
//
#include <hip/hip_runtime.h>
#include <math.h>

// Problem constants (match reference): B=512, T=200, Q=1024
constexpr int BB   = 512;
constexpr int TT   = 200;
constexpr int QQ   = 1024;
constexpr int TM1  = TT - 1;            // 199
constexpr int NPAIR = BB * TM1;         // 101,888 (b,t) pairs
constexpr int WAVES_PER_BLOCK = 8;      // 256 threads, wave32
constexpr int NBLOCKS = NPAIR / WAVES_PER_BLOCK; // 12,736 (exact)

typedef __attribute__((ext_vector_type(4))) float f32x4;
typedef __attribute__((ext_vector_type(2))) float v2f;
typedef __attribute__((ext_vector_type(8))) float v8f;

// Sum of one float per lane across the wave (32 lanes), using the f32 WMMA.
// B = all-ones is layout-invariant; A uses the documented 16x4 f32 layout:
//   lanes 0-15: VGPR0 -> A[m][0], VGPR1 -> A[m][1];  lanes 16-31: K=2,3.
// With a = {v, 0}: A[m][0]=v[m], A[m][2]=v[m+16], rest 0.
// D[m][n] = v[m] + v[m+16] for all n (documented D layout: VGPR j, lanes 0-15
// hold row M=j, lanes 16-31 hold row M=j+8). So per lane, sum_j d[j] is the
// half-wave pair-sum total; adding the opposite half's total finishes it.
__device__ __forceinline__ float wave_sum_wmma(float v) {
    v2f a; a.x = v;    a.y = 0.0f;
    v2f b; b.x = 1.0f; b.y = 1.0f;
    v8f c = {};
    c = __builtin_amdgcn_wmma_f32_16x16x4_f32(
        /*neg_a=*/false, a, /*neg_b=*/false, b,
        /*c_mod=*/(short)0, c, /*reuse_a=*/false, /*reuse_b=*/false);
    float s = ((c[0] + c[1]) + (c[2] + c[3])) + ((c[4] + c[5]) + (c[6] + c[7]));
    s += __shfl_xor(s, 16, 32);
    return s;
}

// One wave per (b,t) pair. Streams pred row (4 KB) + batch row (8 KB) once,
// 128-bit non-temporal coalesced loads, computes p, a, bce.
__global__ __launch_bounds__(256) void pair_kernel(
    const float* __restrict__ pred,   // [B, T, Q]
    const float* __restrict__ batch,  // [B, T, 2Q]
    float* __restrict__ out,          // [1 + N + N]: loss, p, a
    float* __restrict__ blk_bce)      // [NBLOCKS] partial bce sums
{
    __shared__ float sm_bce[WAVES_PER_BLOCK];

    const int wave = threadIdx.x >> 5;
    const int lane = threadIdx.x & 31;
    const int idx  = blockIdx.x * WAVES_PER_BLOCK + wave;   // < NPAIR (exact)
    const int b    = idx / TM1;
    const int t    = idx - b * TM1 + 1;                     // 1..T-1

    const f32x4* predRow = (const f32x4*)(pred  + (size_t)(b * TT + t - 1) * QQ);
    const f32x4* corRow  = (const f32x4*)(batch + (size_t)(b * TT + t) * (2 * QQ));
    const f32x4* incRow  = corRow + (QQ / 4);

    float dot = 0.0f, diff = 0.0f;
#pragma unroll
    for (int k = 0; k < QQ / 128; ++k) {        // 8 iterations, fully unrolled
        const int o = k * 32 + lane;            // float4 index within row
        f32x4 pv = __builtin_nontemporal_load(predRow + o);
        f32x4 cv = __builtin_nontemporal_load(corRow + o);
        f32x4 iv = __builtin_nontemporal_load(incRow + o);
        dot  += pv.x * (cv.x + iv.x) + pv.y * (cv.y + iv.y)
              + pv.z * (cv.z + iv.z) + pv.w * (cv.w + iv.w);
        diff += (cv.x - iv.x) + (cv.y - iv.y) + (cv.z - iv.z) + (cv.w - iv.w);
    }

    // Wave-wide sums via f32 WMMA (EXEC is all-ones here: no divergence above).
    const float p = wave_sum_wmma(dot);
    const float s = wave_sum_wmma(diff);

    const float a    = floorf((s + 1.0f) * 0.5f);
    const float logp = fmaxf(logf(p), -100.0f);
    const float l1mp = fmaxf(log1pf(-p), -100.0f);
    const float bce  = -(a * logp + (1.0f - a) * l1mp);

    if (lane == 0) {
        out[1 + idx]         = p;   // prediction
        out[1 + NPAIR + idx] = a;   // ground truth
        sm_bce[wave]         = bce;
    }
    __syncthreads();
    if (threadIdx.x == 0) {
        float acc = 0.0f;
#pragma unroll
        for (int j = 0; j < WAVES_PER_BLOCK; ++j) acc += sm_bce[j];
        blk_bce[blockIdx.x] = acc;
    }
}

// Deterministic final reduction: fixed per-thread sequential sums + fixed tree.
__global__ __launch_bounds__(256) void loss_kernel(
    const float* __restrict__ blk_bce, float* __restrict__ out)
{
    __shared__ float sm[256];
    float s = 0.0f;
    for (int i = threadIdx.x; i < NBLOCKS; i += 256) s += blk_bce[i];
    sm[threadIdx.x] = s;
    __syncthreads();
    for (int off = 128; off > 0; off >>= 1) {
        if (threadIdx.x < (unsigned)off) sm[threadIdx.x] += sm[threadIdx.x + off];
        __syncthreads();
    }
    if (threadIdx.x == 0) out[0] = sm[0] * (1.0f / (float)TM1);
}

extern "C" void kernel_launch(void* const* d_in, const int* in_sizes, int n_in,
                              void* d_out, int out_size, void* d_ws, size_t ws_size,
                              hipStream_t stream) {
    const float* pred  = (const float*)d_in[0];   // [512,200,1024] f32
    const float* batch = (const float*)d_in[1];   // [512,200,2048] f32
    float* out = (float*)d_out;                   // [1 + 101888 + 101888] f32
    float* blk = (float*)d_ws;                    // NBLOCKS*4 = ~51 KB scratch

    pair_kernel<<<NBLOCKS, 256, 0, stream>>>(pred, batch, out, blk);
    loss_kernel<<<1, 256, 0, stream>>>(blk, out);
}